// GCF_63883343560804
// MI455X (gfx1250) — compile-verified
//
#include <hip/hip_runtime.h>
#include <hip/hip_bf16.h>

// ---------------- problem constants (from reference) ----------------
#define USER_NUM  20000
#define ITEM_NUM  30000
#define N_MAIN    50000
#define N_UL      40000
#define N_IL      60000
#define EMBED     80
#define BATCH     16384
#define NNZ_MAIN  2000000LL
#define NNZ_UL    1000000LL
#define NNZ_IL    1500000LL
#define SLOPE     0.01f
#define NORM_EPS  1e-12f

typedef __attribute__((ext_vector_type(2))) float v2f;
typedef __attribute__((ext_vector_type(8))) float v8f;

__device__ __forceinline__ float leaky(float x) { return x > 0.0f ? x : SLOPE * x; }

// ---------------- build x = [a ; b], also write into concat col block 0 ----
__global__ void k_build_x(const float* __restrict__ a, const float* __restrict__ b,
                          float* __restrict__ x, float* __restrict__ cat,
                          int na, int n, int F, int catStride) {
    int t = blockIdx.x * blockDim.x + threadIdx.x;
    int total = n * F;
    if (t >= total) return;
    int r = t / F, c = t - r * F;
    float v = (r < na) ? a[(long long)r * F + c] : b[(long long)(r - na) * F + c];
    x[t] = v;
    cat[(long long)r * catStride + c] = v;
}

// ---------------- s1 = x  (folds the +I term), s2 = 0 ----------------------
__global__ void k_init_s(const float* __restrict__ x, float* __restrict__ s1,
                         float* __restrict__ s2, int total) {
    int t = blockIdx.x * blockDim.x + threadIdx.x;
    if (t >= total) return;
    s1[t] = x[t];
    s2[t] = 0.0f;
}

// ---------------- COO dual SpMM: s1 += L@x, s2 += L@(x*x) ------------------
// one thread = one (edge, 4-feature chunk); atomics hit L2
__global__ void k_spmm_dual(const int* __restrict__ row, const int* __restrict__ col,
                            const float* __restrict__ val, const float* __restrict__ x,
                            float* __restrict__ s1, float* __restrict__ s2,
                            long long nnz, int F) {
    long long t = (long long)blockIdx.x * blockDim.x + threadIdx.x;
    const int C = F >> 2;
    long long total = nnz * C;
    if (t >= total) return;
    long long e = t / C;
    int c = (int)(t - e * C);
    int rr = row[e], cc = col[e];
    float v = val[e];
    if (e + 64 < nnz) __builtin_prefetch(&val[e + 64], 0, 3);   // global_prefetch_b8
    float4 xv = ((const float4*)(x + (long long)cc * F))[c];
    float* p1 = s1 + (long long)rr * F + c * 4;
    float* p2 = s2 + (long long)rr * F + c * 4;
    atomicAdd(p1 + 0, v * xv.x);
    atomicAdd(p1 + 1, v * xv.y);
    atomicAdd(p1 + 2, v * xv.z);
    atomicAdd(p1 + 3, v * xv.w);
    atomicAdd(p2 + 0, v * xv.x * xv.x);
    atomicAdd(p2 + 1, v * xv.y * xv.y);
    atomicAdd(p2 + 2, v * xv.z * xv.z);
    atomicAdd(p2 + 3, v * xv.w * xv.w);
}

// ---------------- fused GNN layer: WMMA fp32 GEMM x2 + bias + leaky +
//                  sum + row-L2-normalize + scatter to x_next and concat ----
// wave32: one wave per 16-row tile; A lane map (16x4 f32): lane l holds
// rows l&15, K = (l>>4)*2 + v ; B (4x16): lane l holds K = (l>>4)*2 + v,
// col l&15 ; C/D: VGPR v -> row v + (l>>4)*8, col l&15.
template <int INF, int OUTF>
__global__ __launch_bounds__(256) void k_gnn_layer(
    const float* __restrict__ s1, const float* __restrict__ s2,
    const float* __restrict__ W1, const float* __restrict__ b1,
    const float* __restrict__ W2, const float* __restrict__ b2,
    float* __restrict__ xn, float* __restrict__ cat,
    int N, int catStride, int colOff) {
    __shared__ float lds[8 * (16 * 80 + 16)];
    const int NT = OUTF / 16;
    int wave = threadIdx.x >> 5, lane = threadIdx.x & 31;
    int tile = blockIdx.x * 8 + wave;
    if (tile * 16 >= N) return;
    float* Y  = lds + wave * (16 * 80 + 16);
    float* RN = Y + 16 * 80;

    int r = lane & 15, kh = lane >> 4;
    long long arow = (long long)(tile * 16 + r) * INF;

    v8f acc1[NT], acc2[NT];
#pragma unroll
    for (int nt = 0; nt < NT; ++nt)
#pragma unroll
        for (int i = 0; i < 8; ++i) { acc1[nt][i] = 0.0f; acc2[nt][i] = 0.0f; }

#pragma unroll 4
    for (int k = 0; k < INF; k += 4) {
        int kk = k + kh * 2;
        v2f a1 = *(const v2f*)(s1 + arow + kk);
        v2f a2 = *(const v2f*)(s2 + arow + kk);
#pragma unroll
        for (int nt = 0; nt < NT; ++nt) {
            int n = nt * 16 + r;
            v2f bb;
            bb.x = W1[(long long)kk * OUTF + n];
            bb.y = W1[(long long)(kk + 1) * OUTF + n];
            acc1[nt] = __builtin_amdgcn_wmma_f32_16x16x4_f32(
                false, a1, false, bb, (short)0, acc1[nt], false, false);
            v2f b2v;
            b2v.x = W2[(long long)kk * OUTF + n];
            b2v.y = W2[(long long)(kk + 1) * OUTF + n];
            acc2[nt] = __builtin_amdgcn_wmma_f32_16x16x4_f32(
                false, a2, false, b2v, (short)0, acc2[nt], false, false);
        }
    }

    // bias + leaky + sum -> LDS tile
#pragma unroll
    for (int nt = 0; nt < NT; ++nt) {
        float bb1 = b1[nt * 16 + r], bb2 = b2[nt * 16 + r];
#pragma unroll
        for (int v = 0; v < 8; ++v) {
            float p1 = leaky(acc1[nt][v] + bb1);
            float p2 = leaky(acc2[nt][v] + bb2);
            Y[(v + kh * 8) * OUTF + nt * 16 + r] = p1 + p2;
        }
    }
    asm volatile("s_wait_dscnt 0" ::: "memory");

    if (lane < 16) {
        float ss = 0.0f;
#pragma unroll 8
        for (int c = 0; c < OUTF; ++c) { float y = Y[lane * OUTF + c]; ss += y * y; }
        RN[lane] = 1.0f / fmaxf(sqrtf(ss), NORM_EPS);
    }
    asm volatile("s_wait_dscnt 0" ::: "memory");

    for (int i = lane; i < 16 * OUTF; i += 32) {
        int rr = i / OUTF, cc = i - rr * OUTF;
        float yv = Y[i] * RN[rr];
        int node = tile * 16 + rr;
        xn[(long long)node * OUTF + cc] = yv;
        cat[(long long)node * catStride + colOff + cc] = yv;
    }
}

// ---------------- gather h = [mcat[u], mcat[i+U], ucat[u], icat[i]] --------
__global__ void k_gather_h(const int* __restrict__ uIdx, const int* __restrict__ iIdx,
                           const float* __restrict__ mcat, const float* __restrict__ ucat,
                           const float* __restrict__ icat, float* __restrict__ h) {
    const int C = 1088 / 4;  // 272 float4 per row
    int t = blockIdx.x * blockDim.x + threadIdx.x;
    if (t >= BATCH * C) return;
    int b = t / C, c = t - b * C;
    int u = uIdx[b], it = iIdx[b];
    float4 v;
    if (c < 56)       v = ((const float4*)(mcat + (long long)u * 224))[c];
    else if (c < 112) v = ((const float4*)(mcat + (long long)(it + USER_NUM) * 224))[c - 56];
    else if (c < 192) v = ((const float4*)(ucat + (long long)u * 320))[c - 112];
    else              v = ((const float4*)(icat + (long long)it * 320))[c - 192];
    ((float4*)(h + (long long)b * 1088))[c] = v;
}

// ---------------- generic WMMA fp32 GEMM + bias + LeakyReLU ----------------
// grid.x covers M/128 (8 waves x 16 rows), grid.y covers N/(16*NT)
template <int NT>
__global__ __launch_bounds__(256) void k_gemm(
    const float* __restrict__ X, const float* __restrict__ W,
    const float* __restrict__ bias, float* __restrict__ Y,
    int M, int K, int N) {
    int wave = threadIdx.x >> 5, lane = threadIdx.x & 31;
    int m0 = (blockIdx.x * 8 + wave) * 16;
    if (m0 >= M) return;
    int n0 = blockIdx.y * (NT * 16);
    int r = lane & 15, kh = lane >> 4;
    const float* Xr = X + (long long)(m0 + r) * K;

    v8f acc[NT];
#pragma unroll
    for (int nt = 0; nt < NT; ++nt)
#pragma unroll
        for (int i = 0; i < 8; ++i) acc[nt][i] = 0.0f;

    for (int k = 0; k < K; k += 4) {
        int kk = k + kh * 2;
        v2f a = *(const v2f*)(Xr + kk);
#pragma unroll
        for (int nt = 0; nt < NT; ++nt) {
            int n = n0 + nt * 16 + r;
            v2f bb;
            bb.x = W[(long long)kk * N + n];
            bb.y = W[(long long)(kk + 1) * N + n];
            acc[nt] = __builtin_amdgcn_wmma_f32_16x16x4_f32(
                false, a, false, bb, (short)0, acc[nt], false, false);
        }
    }
#pragma unroll
    for (int nt = 0; nt < NT; ++nt) {
        float bb = bias[n0 + nt * 16 + r];
#pragma unroll
        for (int v = 0; v < 8; ++v) {
            float p = leaky(acc[nt][v] + bb);
            Y[(long long)(m0 + v + kh * 8) * N + n0 + nt * 16 + r] = p;
        }
    }
}

// ---------------- last trans layer: 32 -> 1 (no activation) ----------------
__global__ void k_final(const float* __restrict__ X, const float* __restrict__ W,
                        const float* __restrict__ bias, float* __restrict__ out, int M) {
    int t = blockIdx.x * blockDim.x + threadIdx.x;
    if (t >= M) return;
    const float* x = X + (long long)t * 32;
    float s = bias[0];
#pragma unroll
    for (int j = 0; j < 32; ++j) s += x[j] * W[j];
    out[t] = s;
}

// ======================== host-side orchestration ==========================
// Input order (jax tree-flatten, dicts by sorted key):
//  0 userIdx  1 itemIdx  2-4 lap row/col/val  5-7 ul  8-10 il
//  params: gnn(2x[W1,W2,b1,b2]) 11..18 ; gnn_il(3x4) 19..30 ; gnn_ul(3x4) 31..42
//  iEmbd 43 ; ilEmbd 44 ; trans (W,b)x5 45..54 ; uEmbd 55 ; ulEmbd 56
extern "C" void kernel_launch(void* const* d_in, const int* in_sizes, int n_in,
                              void* d_out, int out_size, void* d_ws, size_t ws_size,
                              hipStream_t stream) {
    const int* userIdx = (const int*)d_in[0];
    const int* itemIdx = (const int*)d_in[1];
    const int* lap_row = (const int*)d_in[2];
    const int* lap_col = (const int*)d_in[3];
    const float* lap_val = (const float*)d_in[4];
    const int* ul_row = (const int*)d_in[5];
    const int* ul_col = (const int*)d_in[6];
    const float* ul_val = (const float*)d_in[7];
    const int* il_row = (const int*)d_in[8];
    const int* il_col = (const int*)d_in[9];
    const float* il_val = (const float*)d_in[10];

    const int P = 11;
    const float* gnnW[2][4];   // main: W1,W2,b1,b2
    for (int L = 0; L < 2; ++L)
        for (int j = 0; j < 4; ++j) gnnW[L][j] = (const float*)d_in[P + L * 4 + j];
    const float* ilW[3][4];
    for (int L = 0; L < 3; ++L)
        for (int j = 0; j < 4; ++j) ilW[L][j] = (const float*)d_in[P + 8 + L * 4 + j];
    const float* ulW[3][4];
    for (int L = 0; L < 3; ++L)
        for (int j = 0; j < 4; ++j) ulW[L][j] = (const float*)d_in[P + 20 + L * 4 + j];
    const float* iEmbd  = (const float*)d_in[43];
    const float* ilEmbd = (const float*)d_in[44];
    const float* transW[5], *transB[5];
    for (int L = 0; L < 5; ++L) {
        transW[L] = (const float*)d_in[45 + 2 * L];
        transB[L] = (const float*)d_in[46 + 2 * L];
    }
    const float* uEmbd  = (const float*)d_in[55];
    const float* ulEmbd = (const float*)d_in[56];

    // ---- workspace carve-up (floats) ----
    float* ws = (float*)d_ws;
    size_t off = 0;
    auto alloc = [&](size_t n) { float* p = ws + off; off += n; return p; };
    float* xA   = alloc((size_t)N_IL * EMBED);
    float* xB   = alloc((size_t)N_IL * EMBED);
    float* s1   = alloc((size_t)N_IL * EMBED);
    float* s2   = alloc((size_t)N_IL * EMBED);
    float* catM = alloc((size_t)N_MAIN * 224);
    float* catU = alloc((size_t)N_UL * 320);
    float* catI = alloc((size_t)N_IL * 320);
    float* h0   = alloc((size_t)BATCH * 1088);
    float* h1   = alloc((size_t)BATCH * 256);
    float* h2   = alloc((size_t)BATCH * 128);
    float* h3   = alloc((size_t)BATCH * 64);
    float* h4   = alloc((size_t)BATCH * 32);
    (void)ws_size; (void)in_sizes; (void)n_in; (void)out_size;

    auto cdiv = [](long long a, long long b) { return (int)((a + b - 1) / b); };

    // ---- one GNN stack ----
    auto run_stack = [&](const int* erow, const int* ecol, const float* eval,
                         long long nnz, int N, int nLayers,
                         const float* (*Wb)[4], const int* outFs,
                         float* cat, int catStride) {
        float* xin = xA;
        float* xout = xB;
        int colOff = EMBED;
        for (int L = 0; L < nLayers; ++L) {
            int outF = outFs[L];
            int total = N * EMBED;
            k_init_s<<<cdiv(total, 256), 256, 0, stream>>>(xin, s1, s2, total);
            long long tt = nnz * (EMBED / 4);
            k_spmm_dual<<<cdiv(tt, 256), 256, 0, stream>>>(
                erow, ecol, eval, xin, s1, s2, nnz, EMBED);
            int blocks = cdiv(N / 16, 8);
            if (outF == 80)
                k_gnn_layer<80, 80><<<blocks, 256, 0, stream>>>(
                    s1, s2, Wb[L][0], Wb[L][2], Wb[L][1], Wb[L][3],
                    xout, cat, N, catStride, colOff);
            else
                k_gnn_layer<80, 64><<<blocks, 256, 0, stream>>>(
                    s1, s2, Wb[L][0], Wb[L][2], Wb[L][1], Wb[L][3],
                    xout, cat, N, catStride, colOff);
            colOff += outF;
            float* tmp = xin; xin = xout; xout = tmp;
        }
    };

    const int mainOut[2] = {80, 64};
    const int locOut[3]  = {80, 80, 80};

    // main graph stack
    k_build_x<<<cdiv((long long)N_MAIN * EMBED, 256), 256, 0, stream>>>(
        uEmbd, iEmbd, xA, catM, USER_NUM, N_MAIN, EMBED, 224);
    run_stack(lap_row, lap_col, lap_val, NNZ_MAIN, N_MAIN, 2, gnnW, mainOut, catM, 224);

    // user-location stack ([ulEmbd ; ulEmbd])
    k_build_x<<<cdiv((long long)N_UL * EMBED, 256), 256, 0, stream>>>(
        ulEmbd, ulEmbd, xA, catU, USER_NUM, N_UL, EMBED, 320);
    run_stack(ul_row, ul_col, ul_val, NNZ_UL, N_UL, 3, ulW, locOut, catU, 320);

    // item-location stack ([ilEmbd ; ilEmbd])
    k_build_x<<<cdiv((long long)N_IL * EMBED, 256), 256, 0, stream>>>(
        ilEmbd, ilEmbd, xA, catI, ITEM_NUM, N_IL, EMBED, 320);
    run_stack(il_row, il_col, il_val, NNZ_IL, N_IL, 3, ilW, locOut, catI, 320);

    // gather per-sample features
    k_gather_h<<<cdiv((long long)BATCH * 272, 256), 256, 0, stream>>>(
        userIdx, itemIdx, catM, catU, catI, h0);

    // trans MLP (WMMA fp32)
    k_gemm<4><<<dim3(BATCH / 128, 4), 256, 0, stream>>>(h0, transW[0], transB[0], h1, BATCH, 1088, 256);
    k_gemm<4><<<dim3(BATCH / 128, 2), 256, 0, stream>>>(h1, transW[1], transB[1], h2, BATCH, 256, 128);
    k_gemm<4><<<dim3(BATCH / 128, 1), 256, 0, stream>>>(h2, transW[2], transB[2], h3, BATCH, 128, 64);
    k_gemm<2><<<dim3(BATCH / 128, 1), 256, 0, stream>>>(h3, transW[3], transB[3], h4, BATCH, 64, 32);
    k_final<<<cdiv(BATCH, 256), 256, 0, stream>>>(h4, transW[4], transB[4], (float*)d_out, BATCH);
}